// ActionPredictor_47107201302767
// MI455X (gfx1250) — compile-verified
//
#include <hip/hip_runtime.h>
#include <math.h>

typedef float v2f __attribute__((ext_vector_type(2)));
typedef float v8f __attribute__((ext_vector_type(8)));

#define NN     50000
#define EE     600000
#define FIN    128
#define HID    8
#define BB     256
#define STEPS  12
#define LAYERS 12
#define NCLS   16
#define NTILES (NN / 16)          // 50000 is an exact multiple of 16 -> 3125 full tiles

__device__ __forceinline__ float sigf(float x) { return 1.0f / (1.0f + __expf(-x)); }

// ---------------------------------------------------------------------------
// Zero-fill scratch (hs, cs, q_star) -- deterministic init every call.
// ---------------------------------------------------------------------------
__global__ void k_zero(float* __restrict__ p, int n) {
    int i = blockIdx.x * blockDim.x + threadIdx.x;
    if (i < n) p[i] = 0.0f;
}

// ---------------------------------------------------------------------------
// Fused projection via WMMA f32 16x16x4:
//   cols 0..7  : x @ W_rel.T   -> xr   (to be edge-aggregated)
//   cols 8..15 : x @ W_root.T  -> h_node init (+ b_rel folded in)
// One wave per 16-node tile. NN % 16 == 0, so tiles are always full:
// NO per-lane guards in the K loop -> no EXEC branching, loads cluster and
// pipeline against the WMMA accumulation chain.
// A-frag layout (16x4 f32): lanes 0-15 hold K=k0,k0+1; lanes 16-31 K=k0+2,k0+3.
// C/D layout: VGPR r -> M=r (lanes 0-15) / M=8+r (lanes 16-31), N=lane&15.
// ---------------------------------------------------------------------------
__global__ void k_proj(const float* __restrict__ x,
                       const float* __restrict__ W_rel,
                       const float* __restrict__ W_root,
                       const float* __restrict__ b_rel,
                       float* __restrict__ xr,
                       float* __restrict__ h_node) {
    int lane = threadIdx.x & 31;
    int wid  = threadIdx.x >> 5;
    int tile = blockIdx.x * 8 + wid;          // 8 waves / block
    if (tile >= NTILES) return;               // wave-uniform tail guard only

    int half = lane >> 4;                     // 0: lanes 0-15, 1: lanes 16-31
    int l15  = lane & 15;
    int row  = tile * 16 + l15;               // always < NN (full tiles)

    // per-lane bases; all loads below become base + immediate offset
    const float* xrow = x + (size_t)row * FIN + half * 2;
    const float* wcol = ((l15 < HID) ? (W_rel + l15 * FIN)
                                     : (W_root + (l15 - HID) * FIN)) + half * 2;

    v8f c = {};
    #pragma unroll
    for (int k0 = 0; k0 < FIN; k0 += 4) {
        v2f a, b;
        a.x = xrow[k0];
        a.y = xrow[k0 + 1];
        b.x = wcol[k0];
        b.y = wcol[k0 + 1];
        c = __builtin_amdgcn_wmma_f32_16x16x4_f32(
                /*neg_a=*/false, a, /*neg_b=*/false, b,
                /*c_mod=*/(short)0, c, /*reuse_a=*/false, /*reuse_b=*/false);
    }

    int node0 = tile * 16 + half * 8;
    if (l15 < HID) {
        #pragma unroll
        for (int r = 0; r < 8; r++)
            xr[(node0 + r) * HID + l15] = c[r];
    } else {
        float br = b_rel[l15 - HID];
        #pragma unroll
        for (int r = 0; r < 8; r++)
            h_node[(node0 + r) * HID + (l15 - HID)] = c[r] + br;
    }
}

// ---------------------------------------------------------------------------
// Edge aggregation AFTER projection (linearity of lin_rel): 8 floats/edge,
// both tables (1.6 MB each) are L2-resident. One (edge,feature) per thread.
// ---------------------------------------------------------------------------
__global__ void k_edge(const int* __restrict__ edge_index,
                       const float* __restrict__ xr,
                       float* __restrict__ h_node) {
    int idx = blockIdx.x * blockDim.x + threadIdx.x;
    int e = idx >> 3;
    int k = idx & 7;
    if (e < EE) {
        int s = edge_index[e];        // row 0: src
        int d = edge_index[EE + e];   // row 1: dst
        atomicAdd(&h_node[d * HID + k], xr[s * HID + k]);
    }
}

// ---------------------------------------------------------------------------
// Graph node ranges from the *sorted* batch vector: starts[b] = lower_bound(b).
// Handles empty graphs naturally.
// ---------------------------------------------------------------------------
__global__ void k_starts(const int* __restrict__ batch, int* __restrict__ starts) {
    int b = blockIdx.x * blockDim.x + threadIdx.x;
    if (b > BB) return;
    if (b == BB) { starts[BB] = NN; return; }
    int lo = 0, hi = NN;
    while (lo < hi) {
        int mid = (lo + hi) >> 1;
        if (batch[mid] < b) lo = mid + 1; else hi = mid;
    }
    starts[b] = lo;
}

// ---------------------------------------------------------------------------
// One Set2Set LSTM step: 12 stacked layers, one graph per thread (B=256).
// Weight accesses are wave-uniform -> scalar loads. Gate order: i,f,g,o.
// ---------------------------------------------------------------------------
__global__ void k_lstm(const float* __restrict__ Wih0,      // [32,16]
                       const float* __restrict__ WihR,      // [11,32,8]
                       const float* __restrict__ Whh,       // [12,32,8]
                       const float* __restrict__ bih,       // [12,32]
                       const float* __restrict__ bhh,       // [12,32]
                       const float* __restrict__ q_star,    // [256,16]
                       float* __restrict__ hs,              // [12,256,8]
                       float* __restrict__ cs,              // [12,256,8]
                       float* __restrict__ q_out) {         // [256,8]
    int b = threadIdx.x;
    float inp[2 * HID];
    #pragma unroll
    for (int j = 0; j < 2 * HID; j++) inp[j] = q_star[b * 2 * HID + j];

    for (int l = 0; l < LAYERS; l++) {
        float hl[HID], cl[HID];
        #pragma unroll
        for (int k = 0; k < HID; k++) {
            hl[k] = hs[(l * BB + b) * HID + k];
            cl[k] = cs[(l * BB + b) * HID + k];
        }
        const float* Wih = (l == 0) ? Wih0 : (WihR + (size_t)(l - 1) * 4 * HID * HID);
        int ld = (l == 0) ? 2 * HID : HID;

        float gi[HID], gf[HID], gg[HID], go[HID];
        for (int g = 0; g < 4 * HID; g++) {
            float acc = bih[l * 4 * HID + g] + bhh[l * 4 * HID + g];
            const float* wr = Wih + g * ld;
            for (int j = 0; j < ld; j++) acc += wr[j] * inp[j];
            const float* wh = Whh + (size_t)(l * 4 * HID + g) * HID;
            #pragma unroll
            for (int j = 0; j < HID; j++) acc += wh[j] * hl[j];
            int ki = g & (HID - 1);
            int which = g >> 3;                 // 0=i 1=f 2=g 3=o
            if      (which == 0) gi[ki] = acc;
            else if (which == 1) gf[ki] = acc;
            else if (which == 2) gg[ki] = acc;
            else                 go[ki] = acc;
        }
        #pragma unroll
        for (int k = 0; k < HID; k++) {
            float cnew = sigf(gf[k]) * cl[k] + sigf(gi[k]) * tanhf(gg[k]);
            float hnew = sigf(go[k]) * tanhf(cnew);
            cs[(l * BB + b) * HID + k] = cnew;
            hs[(l * BB + b) * HID + k] = hnew;
            inp[k] = hnew;
        }
    }
    #pragma unroll
    for (int k = 0; k < HID; k++) q_out[b * HID + k] = inp[k];
}

// ---------------------------------------------------------------------------
// Per-graph attention softmax + weighted readout. One block per graph (sorted
// batch -> contiguous node range). Two passes; e values staged in scratch.
// h_node (1.6 MB) stays L2-resident across all 12 steps. Writes
// q_star = concat(q, r).
// ---------------------------------------------------------------------------
__global__ void k_attn(const float* __restrict__ h_node,
                       const int* __restrict__ starts,
                       const float* __restrict__ q,       // [256,8]
                       float* __restrict__ q_star,        // [256,16]
                       float* __restrict__ e_scratch) {   // [N]
    __shared__ float red[256];
    __shared__ float a9[9][256];

    int b   = blockIdx.x;
    int tid = threadIdx.x;
    int s0  = starts[b];
    int s1  = starts[b + 1];

    float qb[HID];
    #pragma unroll
    for (int k = 0; k < HID; k++) qb[k] = q[b * HID + k];

    // pass 1: e_n = <h_node[n], q_b>, track max
    float lmax = -INFINITY;
    for (int n = s0 + tid; n < s1; n += 256) {
        float e = 0.0f;
        #pragma unroll
        for (int k = 0; k < HID; k++) e += h_node[n * HID + k] * qb[k];
        e_scratch[n] = e;
        lmax = fmaxf(lmax, e);
    }
    red[tid] = lmax;
    __syncthreads();
    for (int st = 128; st > 0; st >>= 1) {
        if (tid < st) red[tid] = fmaxf(red[tid], red[tid + st]);
        __syncthreads();
    }
    float gmax = red[0];
    __syncthreads();

    // pass 2: softmax weights + weighted feature sum
    float sw = 0.0f, acc[HID];
    #pragma unroll
    for (int k = 0; k < HID; k++) acc[k] = 0.0f;
    for (int n = s0 + tid; n < s1; n += 256) {
        float w = __expf(e_scratch[n] - gmax);
        sw += w;
        #pragma unroll
        for (int k = 0; k < HID; k++) acc[k] += w * h_node[n * HID + k];
    }
    a9[0][tid] = sw;
    #pragma unroll
    for (int k = 0; k < HID; k++) a9[k + 1][tid] = acc[k];
    __syncthreads();
    for (int st = 128; st > 0; st >>= 1) {
        if (tid < st) {
            #pragma unroll
            for (int j = 0; j < 9; j++) a9[j][tid] += a9[j][tid + st];
        }
        __syncthreads();
    }
    if (tid == 0) {
        float denom = a9[0][0];
        #pragma unroll
        for (int k = 0; k < HID; k++) {
            float r = (denom > 0.0f) ? (a9[k + 1][0] / denom) : 0.0f;
            q_star[b * 16 + k]       = qb[k];
            q_star[b * 16 + HID + k] = r;
        }
    }
}

// ---------------------------------------------------------------------------
// Head: relu(q_star@W1.T+b1) @ W2.T + b2, log_softmax. One graph per thread.
// ---------------------------------------------------------------------------
__global__ void k_head(const float* __restrict__ q_star,
                       const float* __restrict__ W1, const float* __restrict__ b1,
                       const float* __restrict__ W2, const float* __restrict__ b2,
                       float* __restrict__ out) {
    int b = threadIdx.x;
    float t1[16];
    #pragma unroll
    for (int o = 0; o < 16; o++) {
        float acc = b1[o];
        #pragma unroll
        for (int j = 0; j < 16; j++) acc += W1[o * 16 + j] * q_star[b * 16 + j];
        t1[o] = fmaxf(acc, 0.0f);
    }
    float t2[NCLS], m = -INFINITY;
    #pragma unroll
    for (int o = 0; o < NCLS; o++) {
        float acc = b2[o];
        #pragma unroll
        for (int j = 0; j < 16; j++) acc += W2[o * 16 + j] * t1[j];
        t2[o] = acc;
        m = fmaxf(m, acc);
    }
    float se = 0.0f;
    #pragma unroll
    for (int o = 0; o < NCLS; o++) se += __expf(t2[o] - m);
    float lse = m + logf(se);
    #pragma unroll
    for (int o = 0; o < NCLS; o++) out[b * NCLS + o] = t2[o] - lse;
}

// ---------------------------------------------------------------------------
static inline size_t align256(size_t x) { return (x + 255) & ~(size_t)255; }

extern "C" void kernel_launch(void* const* d_in, const int* in_sizes, int n_in,
                              void* d_out, int out_size, void* d_ws, size_t ws_size,
                              hipStream_t stream) {
    const float* x         = (const float*)d_in[0];
    const int*   edge_idx  = (const int*)  d_in[1];
    const int*   batch     = (const int*)  d_in[2];
    const float* W_rel     = (const float*)d_in[3];
    const float* b_rel     = (const float*)d_in[4];
    const float* W_root    = (const float*)d_in[5];
    const float* Wih0      = (const float*)d_in[6];
    const float* WihR      = (const float*)d_in[7];
    const float* Whh       = (const float*)d_in[8];
    const float* bih       = (const float*)d_in[9];
    const float* bhh       = (const float*)d_in[10];
    const float* W1        = (const float*)d_in[11];
    const float* b1        = (const float*)d_in[12];
    const float* W2        = (const float*)d_in[13];
    const float* b2        = (const float*)d_in[14];
    float* out             = (float*)d_out;

    // workspace carve-up (all 256B aligned)
    char* base = (char*)d_ws;
    size_t off = 0;
    float* xr      = (float*)(base + off); off += align256((size_t)NN * HID * 4);
    float* h_node  = (float*)(base + off); off += align256((size_t)NN * HID * 4);
    float* e_scr   = (float*)(base + off); off += align256((size_t)NN * 4);
    int*   starts  = (int*)  (base + off); off += align256((size_t)(BB + 1) * 4);
    float* hs      = (float*)(base + off); off += align256((size_t)LAYERS * BB * HID * 4);
    float* cs      = (float*)(base + off); off += align256((size_t)LAYERS * BB * HID * 4);
    float* q_star  = (float*)(base + off); off += align256((size_t)BB * 2 * HID * 4);
    float* q       = (float*)(base + off); off += align256((size_t)BB * HID * 4);
    (void)ws_size; (void)n_in; (void)in_sizes; (void)out_size;

    // hs, cs, q_star are contiguous (sizes are 256B multiples) -> single zero pass
    {
        int nz = 2 * LAYERS * BB * HID + BB * 2 * HID;
        k_zero<<<(nz + 255) / 256, 256, 0, stream>>>(hs, nz);
    }

    // WMMA projection: xr = x@W_rel.T ; h_node = x@W_root.T + b_rel
    {
        int blocks = (NTILES + 7) / 8;         // 8 waves (tiles) per 256-thread block
        k_proj<<<blocks, 256, 0, stream>>>(x, W_rel, W_root, b_rel, xr, h_node);
    }

    // Edge aggregation: h_node[dst] += xr[src]   (lin_rel pushed through the sum)
    {
        long threads = (long)EE * HID;
        k_edge<<<(int)((threads + 255) / 256), 256, 0, stream>>>(edge_idx, xr, h_node);
    }

    // Graph boundaries
    k_starts<<<(BB + 1 + 63) / 64, 64, 0, stream>>>(batch, starts);

    // Set2Set: 12 sequential (LSTM, attention) steps
    for (int s = 0; s < STEPS; s++) {
        k_lstm<<<1, BB, 0, stream>>>(Wih0, WihR, Whh, bih, bhh, q_star, hs, cs, q);
        k_attn<<<BB, 256, 0, stream>>>(h_node, starts, q, q_star, e_scr);
    }

    // Head MLP + log_softmax
    k_head<<<1, BB, 0, stream>>>(q_star, W1, b1, W2, b2, out);
}